// LightGCN_71571335020552
// MI455X (gfx1250) — compile-verified
//
#include <hip/hip_runtime.h>
#include <math.h>

#ifndef __has_builtin
#define __has_builtin(x) 0
#endif

namespace {

constexpr int kEdges  = 2000000;
constexpr int kUsers  = 200000;
constexpr int kItems  = 100000;
constexpr int kNodes  = kUsers + kItems;   // 300000
constexpr int kDim    = 64;
constexpr int kVec    = kDim / 4;          // float4 per node row = 16
constexpr int kNodeV4 = kNodes * kVec;     // 4.8M float4 elements
constexpr int kBlock  = 256;               // 8 waves (wave32)

constexpr int kTile          = 128;              // edges per wave-tile (512B per array -> 1 async b128 op)
constexpr int kWavesPerBlock = kBlock / 32;      // 8
constexpr int kNumTiles      = kEdges / kTile;   // 15625 (exact)
constexpr int kScatterBlocks = 1024;             // 8192 waves grid-striding over 15625 tiles
static_assert(kEdges % kTile == 0, "no tail tile");

// Pointer types matching the async-to-LDS builtin signature:
// (v4i AS(1)*, v4i AS(3)*, imm offset, imm cpol).
typedef int v4i __attribute__((vector_size(16)));
typedef __attribute__((address_space(1))) v4i* gv4i_p;   // global
typedef __attribute__((address_space(3))) v4i* lv4i_p;   // LDS

__device__ __forceinline__ void atomic_add_f32(float* p, float v) {
  // relaxed, device(agent) scope -> no-return global_atomic_add_f32
  __hip_atomic_fetch_add(p, v, __ATOMIC_RELAXED, __HIP_MEMORY_SCOPE_AGENT);
}

__device__ __forceinline__ void wait_asynccnt_3() {
#if __has_builtin(__builtin_amdgcn_s_wait_asynccnt)
  __builtin_amdgcn_s_wait_asynccnt(3);
#else
  asm volatile("s_wait_asynccnt 0x3" ::: "memory");
#endif
}
__device__ __forceinline__ void wait_asynccnt_0() {
#if __has_builtin(__builtin_amdgcn_s_wait_asynccnt)
  __builtin_amdgcn_s_wait_asynccnt(0);
#else
  asm volatile("s_wait_asynccnt 0x0" ::: "memory");
#endif
}

// ---------------------------------------------------------------------------
// HOT KERNEL (defined first so the disasm snippet shows it).
// Scatter: each wave grid-strides over 128-edge tiles. Edge stream staged into
// per-wave LDS with async-to-LDS DMA (double buffered, ASYNCcnt-synchronized).
// Per wave-iteration: 2 edges x 16 lanes x float4 (one full 256B row gather per
// edge), then 4 no-return f32 atomics per lane into h_next[col].
__global__ void scatter_kernel(const int* __restrict__ row, const int* __restrict__ col,
                               const float* __restrict__ nrm,
                               const float* __restrict__ hprev,
                               float* __restrict__ hnext) {
  const int tid  = threadIdx.x;
  const int wave = tid >> 5;
  const int lane = tid & 31;
  const int half = lane >> 4;   // which edge of the pair
  const int sub  = lane & 15;   // float4 slot within the 64-dim row

  const float4* __restrict__ hp4 = reinterpret_cast<const float4*>(hprev);

  int tile = blockIdx.x * kWavesPerBlock + wave;
  const int stride = gridDim.x * kWavesPerBlock;
  if (tile >= kNumTiles) return;  // wave-uniform

#if defined(__HIP_DEVICE_COMPILE__) && \
    __has_builtin(__builtin_amdgcn_global_load_async_to_lds_b128)
  // [wave][buf][ rows(128) | cols(128) | norms(128) ]  -> 24 KB LDS
  __shared__ int lds[kWavesPerBlock][2][3 * kTile];

  auto stage = [&](int t, int buf) {
    int* l = &lds[wave][buf][0];
    const int base = t * kTile;
    __builtin_amdgcn_global_load_async_to_lds_b128(
        (gv4i_p)(row + base + lane * 4),
        (lv4i_p)(l + lane * 4), 0, 0);
    __builtin_amdgcn_global_load_async_to_lds_b128(
        (gv4i_p)(col + base + lane * 4),
        (lv4i_p)(l + kTile + lane * 4), 0, 0);
    __builtin_amdgcn_global_load_async_to_lds_b128(
        (gv4i_p)(nrm + base + lane * 4),
        (lv4i_p)(l + 2 * kTile + lane * 4), 0, 0);
  };

  stage(tile, 0);
  int buf = 0;
  for (; tile < kNumTiles; tile += stride) {
    const int next = tile + stride;
    if (next < kNumTiles) {
      stage(next, buf ^ 1);
      wait_asynccnt_3();   // current buffer's 3 DMA ops have completed
    } else {
      wait_asynccnt_0();
    }
    const int*   lr = &lds[wave][buf][0];
    const int*   lc = lr + kTile;
    const float* lw = (const float*)(lr + 2 * kTile);
    #pragma unroll 4
    for (int i = 0; i < kTile / 2; ++i) {
      const int   el = 2 * i + half;
      const int   r  = lr[el];
      const int   c  = lc[el];
      const float w  = lw[el];
      // 32-bit indices: r*16+sub < 4.8M, c*64+sub*4 < 19.2M -> GVS addressing
      const float4 v = hp4[r * kVec + sub];
      float* dst = hnext + (c * kDim + sub * 4);
      atomic_add_f32(dst + 0, v.x * w);
      atomic_add_f32(dst + 1, v.y * w);
      atomic_add_f32(dst + 2, v.z * w);
      atomic_add_f32(dst + 3, v.w * w);
    }
    buf ^= 1;
  }
#else
  // Fallback: direct global loads of the edge stream (broadcast within 16 lanes)
  for (; tile < kNumTiles; tile += stride) {
    const int base = tile * kTile;
    #pragma unroll 4
    for (int i = 0; i < kTile / 2; ++i) {
      const int   el = base + 2 * i + half;
      const int   r  = __builtin_nontemporal_load(row + el);
      const int   c  = __builtin_nontemporal_load(col + el);
      const float w  = __builtin_nontemporal_load(nrm + el);
      const float4 v = hp4[r * kVec + sub];
      float* dst = hnext + (c * kDim + sub * 4);
      atomic_add_f32(dst + 0, v.x * w);
      atomic_add_f32(dst + 1, v.y * w);
      atomic_add_f32(dst + 2, v.z * w);
      atomic_add_f32(dst + 3, v.w * w);
    }
  }
#endif
}

// ---------------------------------------------------------------------------
// acc = h_prev = concat(user, item); deg = 0
__global__ void init_x_kernel(const float* __restrict__ user,
                              const float* __restrict__ item,
                              float* __restrict__ hprev,
                              float* __restrict__ acc,
                              float* __restrict__ deg) {
  int i = blockIdx.x * blockDim.x + threadIdx.x;
  if (i < kNodeV4) {
    int node = i >> 4;  // 16 float4 per node
    float4 v;
    if (node < kUsers) {
      v = reinterpret_cast<const float4*>(user)[i];
    } else {
      v = reinterpret_cast<const float4*>(item)[i - kUsers * kVec];
    }
    reinterpret_cast<float4*>(hprev)[i] = v;
    reinterpret_cast<float4*>(acc)[i]   = v;
  }
  if (i < kNodes) deg[i] = 0.0f;
}

// deg[col[e]] += 1
__global__ void degree_kernel(const int* __restrict__ col, float* __restrict__ deg) {
  int e = blockIdx.x * blockDim.x + threadIdx.x;
  if (e < kEdges) {
    int c = __builtin_nontemporal_load(col + e);
    atomic_add_f32(deg + c, 1.0f);
  }
}

// deg[i] -> deg>0 ? deg^-0.5 : 0   (in place)
__global__ void dinv_kernel(float* __restrict__ deg) {
  int i = blockIdx.x * blockDim.x + threadIdx.x;
  if (i < kNodes) {
    float d = deg[i];
    deg[i] = (d > 0.0f) ? (1.0f / sqrtf(d)) : 0.0f;
  }
}

// norm[e] = dinv[row[e]] * dinv[col[e]]
__global__ void norm_kernel(const int* __restrict__ row, const int* __restrict__ col,
                            const float* __restrict__ dinv, float* __restrict__ nrm) {
  int e = blockIdx.x * blockDim.x + threadIdx.x;
  if (e < kEdges) {
    if ((e & 255) == 0 && e + 8192 < kEdges) {     // gfx1250 global_prefetch_b8
      __builtin_prefetch(row + e + 8192, 0, 0);
      __builtin_prefetch(col + e + 8192, 0, 0);
    }
    int r = __builtin_nontemporal_load(row + e);
    int c = __builtin_nontemporal_load(col + e);
    float v = dinv[r] * dinv[c];
    __builtin_nontemporal_store(v, nrm + e);
  }
}

__global__ void zero4_kernel(float* __restrict__ p) {
  int i = blockIdx.x * blockDim.x + threadIdx.x;
  if (i < kNodeV4) {
    reinterpret_cast<float4*>(p)[i] = make_float4(0.f, 0.f, 0.f, 0.f);
  }
}

// acc = (acc + h) * scale   (scale = 1, except 0.25 on the last layer)
__global__ void acc_kernel(float* __restrict__ acc, const float* __restrict__ h, float scale) {
  int i = blockIdx.x * blockDim.x + threadIdx.x;
  if (i < kNodeV4) {
    float4 a = reinterpret_cast<const float4*>(acc)[i];
    float4 b = reinterpret_cast<const float4*>(h)[i];
    float4 r = make_float4((a.x + b.x) * scale, (a.y + b.y) * scale,
                           (a.z + b.z) * scale, (a.w + b.w) * scale);
    reinterpret_cast<float4*>(acc)[i] = r;
  }
}

}  // namespace

extern "C" void kernel_launch(void* const* d_in, const int* in_sizes, int n_in,
                              void* d_out, int out_size, void* d_ws, size_t ws_size,
                              hipStream_t stream) {
  const int*   edge = (const int*)d_in[0];      // [2, E] row-major
  const int*   row  = edge;                     // E
  const int*   col  = edge + kEdges;            // E
  const float* user = (const float*)d_in[1];    // [kUsers, 64]
  const float* item = (const float*)d_in[2];    // [kItems, 64]
  float*       acc  = (float*)d_out;            // [kNodes, 64]

  // workspace layout (floats): h0 | h1 | deg/dinv | norm  (~164 MB)
  float* ws  = (float*)d_ws;
  float* h0  = ws;
  float* h1  = h0 + (size_t)kNodes * kDim;
  float* deg = h1 + (size_t)kNodes * kDim;
  float* nrm = deg + kNodes;

  const int gridNode4 = (kNodeV4 + kBlock - 1) / kBlock;
  const int gridNodes = (kNodes + kBlock - 1) / kBlock;
  const int gridEdges = (kEdges + kBlock - 1) / kBlock;

  init_x_kernel<<<gridNode4, kBlock, 0, stream>>>(user, item, h0, acc, deg);
  degree_kernel<<<gridEdges, kBlock, 0, stream>>>(col, deg);
  dinv_kernel<<<gridNodes, kBlock, 0, stream>>>(deg);
  norm_kernel<<<gridEdges, kBlock, 0, stream>>>(row, col, deg, nrm);

  float* hp = h0;
  float* hn = h1;
  for (int layer = 0; layer < 3; ++layer) {
    zero4_kernel<<<gridNode4, kBlock, 0, stream>>>(hn);
    scatter_kernel<<<kScatterBlocks, kBlock, 0, stream>>>(row, col, nrm, hp, hn);
    const float s = (layer == 2) ? 0.25f : 1.0f;
    acc_kernel<<<gridNode4, kBlock, 0, stream>>>(acc, hn, s);
    float* t = hp; hp = hn; hn = t;
  }
}